// GATModel_31026843746681
// MI455X (gfx1250) — compile-verified
//
#include <hip/hip_runtime.h>
#include <math.h>

#define NN    4096
#define FIN   1024
#define HIDD  64
#define NHEAD 8
#define NCLS  40
#define CP    48   // padded class dim

typedef float v2f __attribute__((ext_vector_type(2)));
typedef float v8f __attribute__((ext_vector_type(8)));

__device__ __forceinline__ float lrelu(float x){ return x >= 0.f ? x : 0.2f*x; }
__device__ __forceinline__ float eluf (float x){ return x > 0.f ? x : expm1f(x); }

// D = A(16x4 f32) * B(4x16 f32) + C(16x16 f32)
__device__ __forceinline__ v8f wmma4(v2f a, v2f b, v8f c) {
  return __builtin_amdgcn_wmma_f32_16x16x4_f32(false, a, false, b, (short)0, c, false, false);
}

// CDNA5 async DMA: copy 16B per lane from global directly into LDS (ASYNCcnt).
// Low 32 bits of a generic shared-memory address == wave-relative LDS byte offset
// (ISA 10.2 aperture rule).
__device__ __forceinline__ void async_b128(float* lds_dst, const float* gsrc) {
  const unsigned loff = (unsigned)(unsigned long long)lds_dst;
  asm volatile("global_load_async_to_lds_b128 %0, %1, off"
               :: "v"(loff), "v"(gsrc) : "memory");
}
__device__ __forceinline__ void wait_async() {
  asm volatile("s_wait_asynccnt 0" ::: "memory");
}

// ---------------------------------------------------------------------------
// K1: Whcat[N][512] = x[N][1024] @ Wcat[1024][512]  (Wcat col o=h*64+hid -> W[h][k][hid])
// grid (256 row-tiles, 4 col-groups of 128), block 256 (8 waves, 16 cols each)
// ---------------------------------------------------------------------------
__global__ __launch_bounds__(256) void k1_gemm_wh(const float* __restrict__ x,
                                                  const float* __restrict__ W,
                                                  float* __restrict__ Whcat) {
  __shared__ float As[16 * 36];    // 16 rows x 32 k (pad 36)
  __shared__ float Bs[32 * 136];   // 32 k x 128 cols (pad 136)
  const int i0   = blockIdx.x * 16;
  const int cg   = blockIdx.y;            // cols cg*128 .. +127 (heads 2cg, 2cg+1)
  const int t    = threadIdx.x;
  const int wave = t >> 5, lane = t & 31;
  const int m16  = lane & 15, hl = lane >> 4;

  v8f acc = {};
  for (int k0 = 0; k0 < FIN; k0 += 32) {
    if (t < 128) {                                   // stage A: 128 x 16B async
      const int mr = t >> 3, q = t & 7;
      async_b128(As + mr * 36 + q * 4, x + (i0 + mr) * FIN + k0 + q * 4);
    }
#pragma unroll
    for (int it = 0; it < 4; ++it) {                 // stage B: 1024 x 16B async
      const int q   = t + it * 256;
      const int kk  = q >> 5;
      const int col = (q & 31) * 4;                  // 0..124
      const int hh  = col >> 6, hid = col & 63;
      async_b128(Bs + kk * 136 + col,
                 W + (2 * cg + hh) * (FIN * HIDD) + (k0 + kk) * HIDD + hid);
    }
    wait_async();
    __syncthreads();
#pragma unroll
    for (int ks = 0; ks < 8; ++ks) {
      const int kb = ks * 4 + 2 * hl;
      v2f a = *(const v2f*)(As + m16 * 36 + kb);
      v2f b;
      b.x = Bs[(kb + 0) * 136 + wave * 16 + m16];
      b.y = Bs[(kb + 1) * 136 + wave * 16 + m16];
      acc = wmma4(a, b, acc);
    }
    __syncthreads();
  }
#pragma unroll
  for (int v = 0; v < 8; ++v) {
    const int row = v + 8 * hl;
    Whcat[(i0 + row) * 512 + cg * 128 + wave * 16 + m16] = acc[v];
  }
}

// ---------------------------------------------------------------------------
// K2: src[h][n] = Wh[h][n][:] . a_src[h][:], same for dst (stored [H][N] and [N][H])
// ---------------------------------------------------------------------------
__global__ void k2_attvec(const float* __restrict__ Whcat,
                          const float* __restrict__ a_src, const float* __restrict__ a_dst,
                          float* __restrict__ srcH, float* __restrict__ dstH,
                          float* __restrict__ dstT) {
  const int idx = blockIdx.x * blockDim.x + threadIdx.x;
  if (idx >= NHEAD * NN) return;
  const int h = idx >> 12, n = idx & (NN - 1);
  const float* row = Whcat + n * 512 + h * HIDD;
  const float* as  = a_src + h * HIDD;
  const float* ad  = a_dst + h * HIDD;
  float s = 0.f, d = 0.f;
#pragma unroll 8
  for (int o = 0; o < HIDD; ++o) { const float w = row[o]; s += w * as[o]; d += w * ad[o]; }
  srcH[h * NN + n] = s;
  dstH[h * NN + n] = d;
  dstT[n * NHEAD + h] = d;
}

// ---------------------------------------------------------------------------
// K3: fused masked-softmax attention + aggregation + elu, all 8 heads per block.
// block = 256 threads (8 waves, wave w == head w), 16 output rows per block.
// Wh tile streams into LDS via async DMA, overlapped with exp() p-computation.
// ---------------------------------------------------------------------------
__global__ __launch_bounds__(256) void k3_layer1(const int* __restrict__ adj,
                                                 const float* __restrict__ Whcat,
                                                 const float* __restrict__ srcH,
                                                 const float* __restrict__ dstH,
                                                 const float* __restrict__ dstT,
                                                 float* __restrict__ hcat) {
  __shared__ unsigned int maskS[16 * 128];  // row bitmask over 4096 j
  __shared__ float redS[256 * 8];           // phase-1 per-thread maxima (8 heads)
  __shared__ float mS[NHEAD * 16];          // row-max score per (head,row)
  __shared__ int   uS[NHEAD * 16];          // uniform (no-neighbor) flag
  __shared__ float WhS[16 * 520];           // Wh chunk: 16 j x 512 cols (pad 520)
  __shared__ float pS[NHEAD * 16 * 17];     // p chunk: [h][row][16 j] (pad 17)
  __shared__ float lS[NHEAD * 16 * 2];      // partial row sums

  const int i0   = blockIdx.x * 16;
  const int t    = threadIdx.x;
  const int wave = t >> 5, lane = t & 31;
  const int m16  = lane & 15, hl = lane >> 4;

  // ---- phase 1: bitmask + per-(row,head) max of dst over neighbors ----
  {
    const int r = t >> 4, g = t & 15;       // 16 rows x 16 groups, 256 j per group
    const int jbase = g * 256;
    const int* arow = adj + (i0 + r) * NN + jbase;
    float mx[NHEAD];
#pragma unroll
    for (int h = 0; h < NHEAD; ++h) mx[h] = -INFINITY;
    for (int w8 = 0; w8 < 8; ++w8) {
      unsigned int bits = 0u;
#pragma unroll
      for (int q = 0; q < 8; ++q) {
        const int4 av = *(const int4*)(arow + w8 * 32 + q * 4);
        const int j = jbase + w8 * 32 + q * 4;
        if (av.x > 0) { bits |= 1u << (q * 4 + 0); const float* p = dstT + (j + 0) * 8;
#pragma unroll
          for (int h = 0; h < 8; ++h) mx[h] = fmaxf(mx[h], p[h]); }
        if (av.y > 0) { bits |= 1u << (q * 4 + 1); const float* p = dstT + (j + 1) * 8;
#pragma unroll
          for (int h = 0; h < 8; ++h) mx[h] = fmaxf(mx[h], p[h]); }
        if (av.z > 0) { bits |= 1u << (q * 4 + 2); const float* p = dstT + (j + 2) * 8;
#pragma unroll
          for (int h = 0; h < 8; ++h) mx[h] = fmaxf(mx[h], p[h]); }
        if (av.w > 0) { bits |= 1u << (q * 4 + 3); const float* p = dstT + (j + 3) * 8;
#pragma unroll
          for (int h = 0; h < 8; ++h) mx[h] = fmaxf(mx[h], p[h]); }
      }
      maskS[r * 128 + g * 8 + w8] = bits;
    }
#pragma unroll
    for (int h = 0; h < NHEAD; ++h) redS[t * 8 + h] = mx[h];
  }
  __syncthreads();
  if (t < 128) {
    const int r = t >> 3, h = t & 7;
    float M = -INFINITY;
    for (int g = 0; g < 16; ++g) M = fmaxf(M, redS[(r * 16 + g) * 8 + h]);
    const int uni = (M == -INFINITY);
    uS[h * 16 + r] = uni;
    mS[h * 16 + r] = uni ? 0.f : lrelu(srcH[h * NN + i0 + r] + M);
  }
  __syncthreads();

  // ---- phase 2: stream j in chunks of 16, WMMA-accumulate ----
  const int ph = t >> 5;                    // p-ownership: head
  const int pr = (t & 31) >> 1;             // row
  const int phalf = t & 1;                  // which 8 j's
  const float psrc = srcH[ph * NN + i0 + pr];
  const float pm   = mS[ph * 16 + pr];
  const int   puni = uS[ph * 16 + pr];
  float lpart = 0.f;

  v8f acc[4] = {};
  for (int j0 = 0; j0 < NN; j0 += 16) {
    __syncthreads();
    // launch async DMA of the Wh chunk (16 x 512 floats, 8 x 16B per thread)
#pragma unroll
    for (int it = 0; it < 8; ++it) {
      const int q  = t + it * 256;
      const int jr = q >> 7, c4 = (q & 127) * 4;
      async_b128(WhS + jr * 520 + c4, Whcat + (j0 + jr) * 512 + c4);
    }
    {                                        // overlap: p values (8 exp per thread)
      const unsigned int mword = maskS[pr * 128 + (j0 >> 5)];
      const int bitbase = (j0 & 31) + phalf * 8;
      const float* drow = dstH + ph * NN + j0 + phalf * 8;
      float* prow = pS + ph * 272 + pr * 17 + phalf * 8;
#pragma unroll
      for (int q = 0; q < 8; ++q) {
        float p;
        if (puni)                               p = 1.f;
        else if ((mword >> (bitbase + q)) & 1u) p = expf(lrelu(psrc + drow[q]) - pm);
        else                                    p = 0.f;
        prow[q] = p;
        lpart += p;
      }
    }
    wait_async();
    __syncthreads();
#pragma unroll
    for (int ks = 0; ks < 4; ++ks) {
      const int kb = ks * 4 + 2 * hl;
      v2f a;
      a.x = pS[wave * 272 + m16 * 17 + kb];
      a.y = pS[wave * 272 + m16 * 17 + kb + 1];
#pragma unroll
      for (int ct = 0; ct < 4; ++ct) {
        v2f b;
        b.x = WhS[(kb + 0) * 520 + wave * 64 + ct * 16 + m16];
        b.y = WhS[(kb + 1) * 520 + wave * 64 + ct * 16 + m16];
        acc[ct] = wmma4(a, b, acc[ct]);
      }
    }
  }
  lS[ph * 32 + pr * 2 + phalf] = lpart;
  __syncthreads();

#pragma unroll
  for (int v = 0; v < 8; ++v) {
    const int rr = v + 8 * hl;
    const float l = lS[wave * 32 + rr * 2] + lS[wave * 32 + rr * 2 + 1];
    const float inv = 1.f / l;
#pragma unroll
    for (int ct = 0; ct < 4; ++ct)
      hcat[(i0 + rr) * 512 + wave * 64 + ct * 16 + m16] = eluf(acc[ct][v] * inv);
  }
}

// ---------------------------------------------------------------------------
// K4: Wh2p[N][48] = hcat[N][512] @ W_out[512][40] (cols 40..47 zero-padded)
// block 128 (waves 0..2 compute one 16-col tile each)
// ---------------------------------------------------------------------------
__global__ __launch_bounds__(128) void k4_gemm_out(const float* __restrict__ hcat,
                                                   const float* __restrict__ W_out,
                                                   float* __restrict__ Wh2p) {
  const int i0 = blockIdx.x * 16;
  const int t = threadIdx.x, wave = t >> 5, lane = t & 31;
  if (wave >= 3) return;
  const int m16 = lane & 15, hl = lane >> 4;
  const int col = wave * 16 + m16;
  v8f acc = {};
  for (int k0 = 0; k0 < 512; k0 += 4) {
    const int kb = k0 + 2 * hl;
    const v2f a = *(const v2f*)(hcat + (i0 + m16) * 512 + kb);
    v2f b;
    b.x = (col < NCLS) ? W_out[(kb + 0) * NCLS + col] : 0.f;
    b.y = (col < NCLS) ? W_out[(kb + 1) * NCLS + col] : 0.f;
    acc = wmma4(a, b, acc);
  }
#pragma unroll
  for (int v = 0; v < 8; ++v)
    Wh2p[(i0 + v + 8 * hl) * CP + col] = acc[v];
}

// ---------------------------------------------------------------------------
// K5: s2[n] = Wh2[n].ao_src ; d2[n] = Wh2[n].ao_dst
// ---------------------------------------------------------------------------
__global__ void k5_attvec2(const float* __restrict__ Wh2p,
                           const float* __restrict__ ao_src, const float* __restrict__ ao_dst,
                           float* __restrict__ s2, float* __restrict__ d2) {
  const int n = blockIdx.x * blockDim.x + threadIdx.x;
  if (n >= NN) return;
  const float* row = Wh2p + n * CP;
  float s = 0.f, d = 0.f;
#pragma unroll 8
  for (int c = 0; c < NCLS; ++c) { const float w = row[c]; s += w * ao_src[c]; d += w * ao_dst[c]; }
  s2[n] = s; d2[n] = d;
}

// ---------------------------------------------------------------------------
// K6: out = log_softmax(elu(att2 @ Wh2)); block 128 (4 waves), 16 rows.
// ---------------------------------------------------------------------------
__global__ __launch_bounds__(128) void k6_layer2(const int* __restrict__ adj,
                                                 const float* __restrict__ Wh2p,
                                                 const float* __restrict__ s2,
                                                 const float* __restrict__ d2,
                                                 float* __restrict__ out) {
  __shared__ unsigned int maskS[16 * 128];
  __shared__ float redS[128];
  __shared__ float mS[16];
  __shared__ int   uS[16];
  __shared__ float WhS[32 * 52];   // Wh2 chunk: 32 j x 48 (pad 52)
  __shared__ float pS[16 * 33];    // p chunk: [row][32 j] (pad 33)
  __shared__ float accS[16 * CP];  // cross-wave accumulator
  __shared__ float lS[128];

  const int i0 = blockIdx.x * 16;
  const int t = threadIdx.x, wave = t >> 5, lane = t & 31;
  const int m16 = lane & 15, hl = lane >> 4;

  for (int q = t; q < 16 * CP; q += 128) accS[q] = 0.f;

  // ---- phase 1 ----
  {
    const int r = t >> 3, g = t & 7;       // 16 rows x 8 groups of 512 j
    const int jbase = g * 512;
    const int* arow = adj + (i0 + r) * NN + jbase;
    float mx = -INFINITY;
    for (int w8 = 0; w8 < 16; ++w8) {
      unsigned int bits = 0u;
#pragma unroll
      for (int q = 0; q < 8; ++q) {
        const int4 av = *(const int4*)(arow + w8 * 32 + q * 4);
        const int j = jbase + w8 * 32 + q * 4;
        if (av.x > 0) { bits |= 1u << (q * 4 + 0); mx = fmaxf(mx, d2[j + 0]); }
        if (av.y > 0) { bits |= 1u << (q * 4 + 1); mx = fmaxf(mx, d2[j + 1]); }
        if (av.z > 0) { bits |= 1u << (q * 4 + 2); mx = fmaxf(mx, d2[j + 2]); }
        if (av.w > 0) { bits |= 1u << (q * 4 + 3); mx = fmaxf(mx, d2[j + 3]); }
      }
      maskS[r * 128 + g * 16 + w8] = bits;
    }
    redS[t] = mx;
  }
  __syncthreads();
  if (t < 16) {
    float M = -INFINITY;
    for (int g = 0; g < 8; ++g) M = fmaxf(M, redS[t * 8 + g]);
    const int uni = (M == -INFINITY);
    uS[t] = uni;
    mS[t] = uni ? 0.f : lrelu(s2[i0 + t] + M);
  }
  __syncthreads();

  // ---- phase 2: chunks of 32 j ----
  const int pr = t >> 3, pseg = t & 7;     // 4 p values per thread
  const float psrc = s2[i0 + pr];
  const float pm   = mS[pr];
  const int   puni = uS[pr];
  float lpart = 0.f;
  v8f acc[3] = {};

  for (int j0 = 0; j0 < NN; j0 += 32) {
    __syncthreads();
    // async DMA of Wh2 chunk (32 x 48 floats, 3 x 16B per thread)
#pragma unroll
    for (int it = 0; it < 3; ++it) {
      const int q  = t + it * 128;
      const int jr = q / 12, c4 = (q % 12) * 4;
      async_b128(WhS + jr * 52 + c4, Wh2p + (j0 + jr) * CP + c4);
    }
    {                                        // overlap: p values
      const unsigned int w = maskS[pr * 128 + (j0 >> 5)];
#pragma unroll
      for (int q = 0; q < 4; ++q) {
        const int jl = pseg * 4 + q;
        float p;
        if (puni)                 p = 1.f;
        else if ((w >> jl) & 1u)  p = expf(lrelu(psrc + d2[j0 + jl]) - pm);
        else                      p = 0.f;
        pS[pr * 33 + jl] = p;
        lpart += p;
      }
    }
    wait_async();
    __syncthreads();
#pragma unroll
    for (int ks = 0; ks < 8; ++ks) {
#pragma unroll
      for (int ct = 0; ct < 3; ++ct) {
        if (((ks * 3 + ct) & 3) != wave) continue;   // wave-uniform split of units
        const int kb = ks * 4 + 2 * hl;
        v2f a;
        a.x = pS[m16 * 33 + kb];
        a.y = pS[m16 * 33 + kb + 1];
        v2f b;
        b.x = WhS[(kb + 0) * 52 + ct * 16 + m16];
        b.y = WhS[(kb + 1) * 52 + ct * 16 + m16];
        acc[ct] = wmma4(a, b, acc[ct]);
      }
    }
  }
  __syncthreads();
#pragma unroll
  for (int ct = 0; ct < 3; ++ct)
#pragma unroll
    for (int v = 0; v < 8; ++v)
      atomicAdd(&accS[(v + 8 * hl) * CP + ct * 16 + m16], acc[ct][v]);
  lS[t] = lpart;
  __syncthreads();

  if (t < 16) {
    float l = 0.f;
    for (int g = 0; g < 8; ++g) l += lS[t * 8 + g];
    const float inv = 1.f / l;
    float mx = -INFINITY;
    for (int c = 0; c < NCLS; ++c) mx = fmaxf(mx, eluf(accS[t * CP + c] * inv));
    float se = 0.f;
    for (int c = 0; c < NCLS; ++c) se += expf(eluf(accS[t * CP + c] * inv) - mx);
    const float lse = mx + logf(se);
    for (int c = 0; c < NCLS; ++c)
      out[(i0 + t) * NCLS + c] = eluf(accS[t * CP + c] * inv) - lse;
  }
}

// ---------------------------------------------------------------------------
extern "C" void kernel_launch(void* const* d_in, const int* in_sizes, int n_in,
                              void* d_out, int out_size, void* d_ws, size_t ws_size,
                              hipStream_t stream) {
  (void)in_sizes; (void)n_in; (void)out_size; (void)ws_size;
  const float* x      = (const float*)d_in[0];
  const int*   adj    = (const int*)  d_in[1];
  const float* W      = (const float*)d_in[2];
  const float* a_src  = (const float*)d_in[3];
  const float* a_dst  = (const float*)d_in[4];
  const float* W_out  = (const float*)d_in[5];
  const float* ao_src = (const float*)d_in[6];
  const float* ao_dst = (const float*)d_in[7];
  float* out = (float*)d_out;

  float* ws     = (float*)d_ws;
  float* Whcat  = ws;                               // 4096*512
  float* srcH   = Whcat + (size_t)NN * 512;         // 8*4096
  float* dstH   = srcH  + (size_t)NHEAD * NN;
  float* dstT   = dstH  + (size_t)NHEAD * NN;       // 4096*8
  float* hcat   = dstT  + (size_t)NN * NHEAD;       // 4096*512
  float* Wh2p   = hcat  + (size_t)NN * 512;         // 4096*48
  float* s2v    = Wh2p  + (size_t)NN * CP;          // 4096
  float* d2v    = s2v   + (size_t)NN;               // 4096

  k1_gemm_wh<<<dim3(NN / 16, 4), 256, 0, stream>>>(x, W, Whcat);
  k2_attvec<<<(NHEAD * NN) / 256, 256, 0, stream>>>(Whcat, a_src, a_dst, srcH, dstH, dstT);
  k3_layer1<<<NN / 16, 256, 0, stream>>>(adj, Whcat, srcH, dstH, dstT, hcat);
  k4_gemm_out<<<NN / 16, 128, 0, stream>>>(hcat, W_out, Wh2p);
  k5_attvec2<<<NN / 128, 128, 0, stream>>>(Wh2p, ao_src, ao_dst, s2v, d2v);
  k6_layer2<<<NN / 16, 128, 0, stream>>>(adj, Wh2p, s2v, d2v, out);
}